// ProposalLoss_58815282152141
// MI455X (gfx1250) — compile-verified
//
#include <hip/hip_runtime.h>
#include <stdint.h>

// ProposalLoss (mip-NeRF 360 lossfun_outer forward), MI455X / gfx1250.
// Memory-bound: ~0.54 GB traffic -> ~23 us at 23.3 TB/s. No matmul -> no WMMA;
// CDNA5 paths used: async global->LDS staging (ASYNCcnt + s_wait_asynccnt),
// wave32 shuffle scan, nontemporal (streaming) cache policy on all traffic.

#define N_ROWS   262144
#define S_BINS   128
#define P_BINS   64
#define T_COLS   (S_BINS + 1)   // 129
#define TH_COLS  (P_BINS + 1)   // 65
#define EPSF     1.1920928955078125e-07f  // FLT_EPSILON

#define WAVES_PER_BLOCK 8
#define THREADS         (WAVES_PER_BLOCK * 32)

// per-wave LDS layout in floats (stride is a multiple of 4 -> 16B alignment)
#define TT_OFF      0     // t row: 129 floats
#define TH_OFF      132   // t_hat row: 65 floats
#define CY_OFF      200   // exclusive prefix sum of w_hat: 65 floats
#define WAVE_STRIDE 272

typedef float v4f __attribute__((ext_vector_type(4)));

static __device__ __forceinline__ unsigned lds_off(const void* p) {
  // Generic pointer to LDS = {shared_aperture_hi32, lds_byte_offset}; the low
  // 32 bits are exactly the VGPR value async-LDS instructions expect.
  return (unsigned)(uintptr_t)p;
}

static __device__ __forceinline__ void async_ld_b32(unsigned lds_addr, const float* gaddr) {
  // LDS address comes from the lds_addr VGPR; tracked by ASYNCcnt. NT: data is streamed
  // exactly once across the dispatch, keep it out of persistent L2 footprint.
  asm volatile("global_load_async_to_lds_b32 %0, %1, off th:TH_LOAD_NT"
               :: "v"(lds_addr), "v"(gaddr)
               : "memory");
}

static __device__ __forceinline__ void wait_asynccnt0() {
  asm volatile("s_wait_asynccnt 0x0" ::: "memory");
}

__global__ __launch_bounds__(THREADS) void proposal_loss_kernel(
    const float* __restrict__ t,      // [N, 129]
    const float* __restrict__ w,      // [N, 128]
    const float* __restrict__ t_hat,  // [N, 65]
    const float* __restrict__ w_hat,  // [N, 64]
    float* __restrict__ out)          // [N, 128]
{
  __shared__ float smem[WAVE_STRIDE * WAVES_PER_BLOCK];

  const int lane = threadIdx.x & 31;
  const int wid  = threadIdx.x >> 5;
  const int row  = blockIdx.x * WAVES_PER_BLOCK + wid;

  float* tt = smem + wid * WAVE_STRIDE + TT_OFF;
  float* th = smem + wid * WAVE_STRIDE + TH_OFF;
  float* cy = smem + wid * WAVE_STRIDE + CY_OFF;

  const float* gt  = t     + (size_t)row * T_COLS;
  const float* gth = t_hat + (size_t)row * TH_COLS;
  const float* gwh = w_hat + (size_t)row * P_BINS;
  const float* gw  = w     + (size_t)row * S_BINS;
  float*       go  = out   + (size_t)row * S_BINS;

  // ---- kick off async global->LDS staging of t (129) and t_hat (65) ----
  // (b32 granularity: row strides 516B/260B are only 4B aligned)
  async_ld_b32(lds_off(&tt[lane]),      gt  + lane);
  async_ld_b32(lds_off(&tt[lane + 32]), gt  + lane + 32);
  async_ld_b32(lds_off(&tt[lane + 64]), gt  + lane + 64);
  async_ld_b32(lds_off(&tt[lane + 96]), gt  + lane + 96);
  async_ld_b32(lds_off(&th[lane]),      gth + lane);
  async_ld_b32(lds_off(&th[lane + 32]), gth + lane + 32);
  if (lane == 0) {
    async_ld_b32(lds_off(&tt[128]), gt  + 128);
    async_ld_b32(lds_off(&th[64]),  gth + 64);
  }

  // ---- overlap: load w_hat, wave32 inclusive scan -> exclusive cy1 in LDS ----
  float a = __builtin_nontemporal_load(gwh + lane);        // w_hat[lane]
  float b = __builtin_nontemporal_load(gwh + lane + 32);   // w_hat[lane+32]
  #pragma unroll
  for (int d = 1; d < 32; d <<= 1) {
    float sa = __shfl_up(a, (unsigned)d, 32);
    float sb = __shfl_up(b, (unsigned)d, 32);
    if (lane >= d) { a += sa; b += sb; }
  }
  b += __shfl(a, 31, 32);          // add total of first half to second half
  cy[1 + lane]  = a;               // cy[k] = sum w_hat[0..k-1], k=1..32
  cy[33 + lane] = b;               // k=33..64
  if (lane == 0) cy[0] = 0.0f;

  // ---- overlap: this lane owns outputs 4*lane .. 4*lane+3 (blocked map,
  //      still fully coalesced across the wave) -> single b128 load of w ----
  const v4f wv = __builtin_nontemporal_load((const v4f*)(gw + (lane << 2)));

  // ---- wait for async LDS tiles (wave-private -> no barrier needed) ----
  wait_asynccnt0();

  // t values for this lane: one ds_load_b128 + one scalar (tt base 16B aligned)
  const v4f t4 = *(const v4f*)(tt + (lane << 2));
  const float t5 = tt[(lane << 2) + 4];

  // ---- 4 outputs per lane: dual branchless binary searches over th[0..64] ----
  v4f res;
  #pragma unroll
  for (int q = 0; q < 4; ++q) {
    const float tl = t4[q];
    const float tr = (q < 3) ? t4[q + 1] : t5;

    // left:  lo  -> #elements of th <  tl   (searchsorted side='left')
    // right: lo2 -> #elements of th <= tr   (searchsorted side='right')
    int lo = 0, hi = TH_COLS;
    int lo2 = 0, hi2 = TH_COLS;
    #pragma unroll
    for (int s = 0; s < 7; ++s) {           // 66 possible results -> 7 steps
      const int mid  = (lo  + hi)  >> 1;
      const int mid2 = (lo2 + hi2) >> 1;
      const float v  = th[mid];
      const float v2 = th[mid2];
      const bool c  = v  <  tl;
      const bool c2 = v2 <= tr;
      lo  = c  ? mid  + 1 : lo;   hi  = c  ? hi  : mid;
      lo2 = c2 ? mid2 + 1 : lo2;  hi2 = c2 ? hi2 : mid2;
    }

    const float w_outer = cy[lo2] - cy[lo];
    const float wi = wv[q];
    float d = wi - w_outer;
    d = fmaxf(d, 0.0f);
    res[q] = d * d / (wi + EPSF);
  }

  // single coalesced b128 streaming store (go + 4*lane is 16B aligned)
  __builtin_nontemporal_store(res, (v4f*)(go + (lane << 2)));
}

extern "C" void kernel_launch(void* const* d_in, const int* in_sizes, int n_in,
                              void* d_out, int out_size, void* d_ws, size_t ws_size,
                              hipStream_t stream) {
  const float* t     = (const float*)d_in[0];  // [N, 129]
  const float* w     = (const float*)d_in[1];  // [N, 128]
  const float* t_hat = (const float*)d_in[2];  // [N, 65]
  const float* w_hat = (const float*)d_in[3];  // [N, 64]
  float* out = (float*)d_out;                  // [N, 128]

  const int blocks = N_ROWS / WAVES_PER_BLOCK; // 32768
  proposal_loss_kernel<<<blocks, THREADS, 0, stream>>>(t, w, t_hat, w_hat, out);
  (void)in_sizes; (void)n_in; (void)out_size; (void)d_ws; (void)ws_size;
}